// MultiHeadAttention_67113158967424
// MI455X (gfx1250) — compile-verified
//
#include <hip/hip_runtime.h>
#include <math.h>

// ---------------------------------------------------------------------------
// MI455X (gfx1250) fused multi-head attention, fp32, WMMA F32 16x16x4 path.
// B=4 T=2048 D=1024 H=16 HD=64. Compute-bound (~103 GFLOP vs ~100MB traffic).
// Data movement uses CDNA5 async tensor path: global_load_async_to_lds_b128
// with double-buffered LDS panels, synchronized via s_wait_asynccnt.
// ---------------------------------------------------------------------------

typedef __attribute__((ext_vector_type(2))) float v2f;
typedef __attribute__((ext_vector_type(8))) float v8f;

#define WMMA_F32(a, b, c) \
    __builtin_amdgcn_wmma_f32_16x16x4_f32(false, (a), false, (b), (short)0, (c), false, false)

static constexpr int Bb = 4;
static constexpr int Tt = 2048;
static constexpr int Dd = 1024;
static constexpr int Hh = 16;
static constexpr int HDc = 64;

// Async 16B copy global -> LDS (CDNA5, ASYNCcnt-tracked). LDS byte address is
// the low 32 bits of the generic pointer (flat LDS addr = {aperture, off32}).
__device__ __forceinline__ void async_copy_b128(const float* __restrict__ gsrc,
                                                float* lds_dst)
{
    unsigned           l = (unsigned)(unsigned long long)(uintptr_t)lds_dst;
    unsigned long long g = (unsigned long long)(uintptr_t)gsrc;
    asm volatile("global_load_async_to_lds_b128 %0, %1, off"
                 :: "v"(l), "v"(g) : "memory");
}
#define WAIT_ASYNCCNT(n) asm volatile("s_wait_asynccnt %0" :: "n"(n) : "memory")

// ---------------------------------------------------------------------------
// GEMM  C[M,N] = A[M,K] * B[N,K]^T   (both operands row-major, K contiguous)
// Block: 256 threads = 8 waves, 128x128 output tile.
// K-panel 16, double-buffered via async loads to LDS (4 async ops / thread
// per panel -> s_wait_asynccnt 4 waits exactly on the current panel).
// LDS rows padded to 20 floats -> conflict-free WMMA fragment gathers.
// ---------------------------------------------------------------------------
__global__ __launch_bounds__(256) void gemm_nt_kernel(
    const float* __restrict__ A, const float* __restrict__ Bm,
    float* __restrict__ C, int M, int N, int K)
{
    constexpr int MB = 128, NB = 128, KP = 16, PAD = 20;
    __shared__ float As[2][MB][PAD];
    __shared__ float Bs[2][NB][PAD];

    const int tid  = threadIdx.x;
    const int lane = tid & 31;
    const int wave = tid >> 5;
    const int l16  = lane & 15;
    const int half = lane >> 4;           // 0: lanes 0-15, 1: lanes 16-31
    const int mblk = blockIdx.y * MB;
    const int nblk = blockIdx.x * NB;

    v8f acc[8];
    #pragma unroll
    for (int nt = 0; nt < 8; ++nt)
        #pragma unroll
        for (int r = 0; r < 8; ++r) acc[nt][r] = 0.0f;

    // Stage one 128x16 panel of A and of B into LDS buffer `buf` (async).
    auto stage = [&](int kp, int buf) {
        #pragma unroll
        for (int i = 0; i < 2; ++i) {
            int lin = i * 256 + tid;      // float4 index 0..511
            int row = lin >> 2;           // 4 float4 per 16-float row
            int c4  = lin & 3;
            async_copy_b128(&A[(size_t)(mblk + row) * K + kp + c4 * 4],
                            &As[buf][row][c4 * 4]);
            async_copy_b128(&Bm[(size_t)(nblk + row) * K + kp + c4 * 4],
                            &Bs[buf][row][c4 * 4]);
        }
    };

    const int P = K / KP;
    stage(0, 0);
    for (int p = 0; p < P; ++p) {
        const int cur = p & 1;
        if (p + 1 < P) {                  // prefetch next panel into other buf
            stage((p + 1) * KP, cur ^ 1);
            WAIT_ASYNCCNT(4);             // current panel's 4 ops retired
        } else {
            WAIT_ASYNCCNT(0);
        }
        __syncthreads();

        #pragma unroll
        for (int kk = 0; kk < KP / 4; ++kk) {
            const int kc = kk * 4 + half * 2;   // frag K columns per half-wave
            v2f a;
            a.x = As[cur][wave * 16 + l16][kc];
            a.y = As[cur][wave * 16 + l16][kc + 1];
            #pragma unroll
            for (int nt = 0; nt < 8; ++nt) {
                v2f b;
                b.x = Bs[cur][nt * 16 + l16][kc];
                b.y = Bs[cur][nt * 16 + l16][kc + 1];
                acc[nt] = WMMA_F32(a, b, acc[nt]);
            }
        }
        __syncthreads();   // all waves done with buf `cur` before it refills
    }

    // C/D layout: VGPR r holds M=r (lanes 0-15) and M=r+8 (lanes 16-31); N=lane%16.
    #pragma unroll
    for (int nt = 0; nt < 8; ++nt)
        #pragma unroll
        for (int r = 0; r < 8; ++r) {
            int m = mblk + wave * 16 + r + half * 8;
            int n = nblk + nt * 16 + l16;
            C[(size_t)m * N + n] = acc[nt][r];
        }
}

// ---------------------------------------------------------------------------
// RMSNorm (over HD=64) + rotary, in place on [B,T,H,HD] fp32.
// One wave32 per row: lane j owns the rotary pair (x[j], x[j+32]).
// ---------------------------------------------------------------------------
__global__ __launch_bounds__(256) void rmsnorm_rope_kernel(float* __restrict__ X, int nrows)
{
    int row  = blockIdx.x * 8 + (threadIdx.x >> 5);
    int lane = threadIdx.x & 31;
    if (row >= nrows) return;
    int t = (row / Hh) % Tt;

    float x1 = X[(size_t)row * HDc + lane];
    float x2 = X[(size_t)row * HDc + lane + 32];

    float ss = x1 * x1 + x2 * x2;
    #pragma unroll
    for (int off = 16; off > 0; off >>= 1) ss += __shfl_xor(ss, off, 32);
    float rn = rsqrtf(ss * (1.0f / 64.0f) + 1e-6f);
    x1 *= rn; x2 *= rn;

    // angular_freq[j] = (1/1024)^(j/15) for j<16, 0 for j>=16 (zero-padded table)
    float freq = (lane < 16) ? __powf(1.0f / 1024.0f, (float)lane * (1.0f / 15.0f)) : 0.0f;
    float th = (float)t * freq;
    float c = cosf(th), s = sinf(th);

    X[(size_t)row * HDc + lane]      =  x1 * c + x2 * s;
    X[(size_t)row * HDc + lane + 32] = -x1 * s + x2 * c;
}

// ---------------------------------------------------------------------------
// Flash attention: one wave per 16-row s-tile of one (b,h). Causal loop over
// t-tiles 0..st. K/V tiles double-buffered via async loads (16 async ops/lane
// per tile -> s_wait_asynccnt 16 waits exactly on the current tile).
// S = Q*K^T via 16 WMMAs, online softmax, O += P*V via 16 WMMAs with the P
// tile re-laid-out through padded LDS.
// ---------------------------------------------------------------------------
__global__ __launch_bounds__(32) void attn_kernel(
    const float* __restrict__ Q, const float* __restrict__ Kx,
    const float* __restrict__ V, float* __restrict__ O)
{
    constexpr int KVP = 68;   // 16x64 tile, row stride 68 -> conflict-free gathers
    constexpr int PP  = 20;   // 16x16 P tile, row stride 20
    __shared__ float Ks[2][16][KVP];
    __shared__ float Vs[2][16][KVP];
    __shared__ float Ps[16][PP];

    const int lane = threadIdx.x;
    const int l16  = lane & 15;
    const int half = lane >> 4;
    const int st   = blockIdx.x;   // s-tile
    const int h    = blockIdx.y;
    const int b    = blockIdx.z;
    const size_t browbase = (size_t)b * Tt;    // token-row base: b*T + t

    // Q fragments for all 16 K-steps (HD=64 / 4).
    v2f qf[16];
    {
        const float* qrow = &Q[((browbase + st * 16 + l16) * Hh + h) * (size_t)HDc];
        #pragma unroll
        for (int kk = 0; kk < 16; ++kk) {
            int kc = kk * 4 + half * 2;
            qf[kk].x = qrow[kc];
            qf[kk].y = qrow[kc + 1];
        }
    }

    float m8[8], l8[8];
    #pragma unroll
    for (int r = 0; r < 8; ++r) { m8[r] = -3.0e38f; l8[r] = 0.0f; }
    v8f oacc[4];
    #pragma unroll
    for (int nt = 0; nt < 4; ++nt)
        #pragma unroll
        for (int r = 0; r < 8; ++r) oacc[nt][r] = 0.0f;

    // Async-stage one 16x64 K tile + V tile into buffer `buf`.
    auto stage_kv = [&](int tt, int buf) {
        #pragma unroll
        for (int i = 0; i < 8; ++i) {
            int idx4 = i * 32 + lane;    // 256 float4 per tile
            int row  = idx4 >> 4;
            int c4   = idx4 & 15;
            size_t g = ((browbase + tt * 16 + row) * Hh + h) * (size_t)HDc + c4 * 4;
            async_copy_b128(&Kx[g], &Ks[buf][row][c4 * 4]);
            async_copy_b128(&V[g],  &Vs[buf][row][c4 * 4]);
        }
    };

    stage_kv(0, 0);
    for (int tt = 0; tt <= st; ++tt) {
        const int cur = tt & 1;
        if (tt < st) {                    // prefetch next t-tile into other buf
            stage_kv(tt + 1, cur ^ 1);
            WAIT_ASYNCCNT(16);            // current tile's 16 ops retired
        } else {
            WAIT_ASYNCCNT(0);
        }
        __syncthreads();   // single-wave WG: S_NOP barrier + compiler fence

        // S tile = Q * K^T (scaled)
        v8f s;
        #pragma unroll
        for (int r = 0; r < 8; ++r) s[r] = 0.0f;
        #pragma unroll
        for (int kk = 0; kk < 16; ++kk) {
            int kc = kk * 4 + half * 2;
            v2f bf;
            bf.x = Ks[cur][l16][kc];
            bf.y = Ks[cur][l16][kc + 1];
            s = WMMA_F32(qf[kk], bf, s);
        }
        #pragma unroll
        for (int r = 0; r < 8; ++r) s[r] *= 0.125f;   // 1/sqrt(64)

        if (tt == st) {  // causal mask on diagonal tile: keep n <= m
            #pragma unroll
            for (int r = 0; r < 8; ++r) {
                int mrow = r + half * 8;
                if (l16 > mrow) s[r] = -3.0e38f;
            }
        }

        // Online softmax per row (rows live in 16-lane halves).
        #pragma unroll
        for (int r = 0; r < 8; ++r) {
            float tmax = s[r];
            #pragma unroll
            for (int off = 1; off < 16; off <<= 1)
                tmax = fmaxf(tmax, __shfl_xor(tmax, off, 32));
            float mnew  = fmaxf(m8[r], tmax);
            float alpha = __expf(m8[r] - mnew);
            float p     = __expf(s[r] - mnew);
            float rsum  = p;
            #pragma unroll
            for (int off = 1; off < 16; off <<= 1)
                rsum += __shfl_xor(rsum, off, 32);
            l8[r] = l8[r] * alpha + rsum;
            m8[r] = mnew;
            #pragma unroll
            for (int nt = 0; nt < 4; ++nt) oacc[nt][r] *= alpha;
            Ps[r + half * 8][l16] = p;   // C-layout -> row-major P tile
        }
        __syncthreads();

        // O += P * V   (K dim = 16 local t, N dim = 64 d in 4 tiles)
        #pragma unroll
        for (int nt = 0; nt < 4; ++nt) {
            #pragma unroll
            for (int kk = 0; kk < 4; ++kk) {
                int kc = kk * 4 + half * 2;
                v2f a, bb;
                a.x  = Ps[l16][kc];
                a.y  = Ps[l16][kc + 1];
                bb.x = Vs[cur][kc][nt * 16 + l16];
                bb.y = Vs[cur][kc + 1][nt * 16 + l16];
                oacc[nt] = WMMA_F32(a, bb, oacc[nt]);
            }
        }
        __syncthreads();
    }

    // Normalize and write [B,T,H,HD].
    #pragma unroll
    for (int nt = 0; nt < 4; ++nt)
        #pragma unroll
        for (int r = 0; r < 8; ++r) {
            int srow = st * 16 + r + half * 8;
            O[((browbase + srow) * Hh + h) * (size_t)HDc + nt * 16 + l16] =
                oacc[nt][r] / l8[r];
        }
}

// ---------------------------------------------------------------------------
extern "C" void kernel_launch(void* const* d_in, const int* in_sizes, int n_in,
                              void* d_out, int out_size, void* d_ws, size_t ws_size,
                              hipStream_t stream)
{
    (void)in_sizes; (void)n_in; (void)out_size; (void)ws_size;
    const float* x  = (const float*)d_in[0];
    const float* wq = (const float*)d_in[1];
    const float* wk = (const float*)d_in[2];
    const float* wv = (const float*)d_in[3];
    const float* wo = (const float*)d_in[4];
    // d_in[5] = causal mask (bool) -- applied analytically, not read.
    float* out = (float*)d_out;

    const size_t MT = (size_t)Bb * Tt;        // 8192 token rows
    float* q  = (float*)d_ws;                 // 4 buffers of 32 MB
    float* k  = q  + MT * Dd;
    float* v  = k  + MT * Dd;
    float* ao = v  + MT * Dd;

    dim3 gblk(Dd / 128, (unsigned)(MT / 128));   // (8, 64)
    gemm_nt_kernel<<<gblk, 256, 0, stream>>>(x, wq, q, (int)MT, Dd, Dd);
    gemm_nt_kernel<<<gblk, 256, 0, stream>>>(x, wk, k, (int)MT, Dd, Dd);
    gemm_nt_kernel<<<gblk, 256, 0, stream>>>(x, wv, v, (int)MT, Dd, Dd);

    int nrows = Bb * Tt * Hh;                 // 131072
    rmsnorm_rope_kernel<<<nrows / 8, 256, 0, stream>>>(q, nrows);
    rmsnorm_rope_kernel<<<nrows / 8, 256, 0, stream>>>(k, nrows);

    dim3 ga(Tt / 16, Hh, Bb);                 // (128, 16, 4), 1 wave per block
    attn_kernel<<<ga, 32, 0, stream>>>(q, k, v, ao);

    gemm_nt_kernel<<<gblk, 256, 0, stream>>>(ao, wo, out, (int)MT, Dd, Dd);
}